// FastFlow3D_27453430956219
// MI455X (gfx1250) — compile-verified
//
#include <hip/hip_runtime.h>
#include <math.h>

#define B_   2
#define N0_  8192
#define N1_  8192
#define K_   32
#define H_   128
#define R2_  1.0f
#define CAP_ 64

typedef float v2f __attribute__((ext_vector_type(2)));
typedef float v4f __attribute__((ext_vector_type(4)));
typedef float v8f __attribute__((ext_vector_type(8)));

// ---------------------------------------------------------------------------
// Preprocess: pack target points as (x,y,z,0) + separate |t|^2, and candidate
// points as (x,y,z,|p|^2).  The zero in the query's K=3 slot annihilates the
// norm riding in the candidate's K=3 slot inside the WMMA dot product.
// ---------------------------------------------------------------------------
__global__ void ff3d_pre(const float* __restrict__ p0,
                         const float* __restrict__ p1,
                         const float* __restrict__ cf,
                         float* __restrict__ targ4,
                         float* __restrict__ p14,
                         float* __restrict__ qn) {
  int i = blockIdx.x * blockDim.x + threadIdx.x;
  if (i >= B_ * N0_) return;   // N0_ == N1_, one index covers both streams

  float tx = p0[i * 3 + 0] + cf[i * 3 + 0];
  float ty = p0[i * 3 + 1] + cf[i * 3 + 1];
  float tz = p0[i * 3 + 2] + cf[i * 3 + 2];
  targ4[i * 4 + 0] = tx;
  targ4[i * 4 + 1] = ty;
  targ4[i * 4 + 2] = tz;
  targ4[i * 4 + 3] = 0.0f;                       // K=3 pad MUST be zero
  qn[i] = tx * tx + ty * ty + tz * tz;

  float px = p1[i * 3 + 0];
  float py = p1[i * 3 + 1];
  float pz = p1[i * 3 + 2];
  p14[i * 4 + 0] = px;
  p14[i * 4 + 1] = py;
  p14[i * 4 + 2] = pz;
  p14[i * 4 + 3] = px * px + py * py + pz * pz;  // norm in K=3 slot (x 0 = ok)
}

// ---------------------------------------------------------------------------
// Main kernel: 128 threads = 4 waves; each wave owns 16 queries.
// Phase 1: software-pipelined WMMA 16x16x4 f32 distance scan over all 8192
//          candidates; radius hits appended to per-query LDS lists
//          (wave-local, no cross-wave sync needed).
// Phase 2: exact 32-nearest select (rare) + 6->128 ReLU MLP + maxpool +
//          131->3 projection with a wave shuffle reduction.
// ---------------------------------------------------------------------------
__global__ __launch_bounds__(128) void ff3d_main(
    const float* __restrict__ targ4, const float* __restrict__ p14,
    const float* __restrict__ qn,
    const float* __restrict__ f0, const float* __restrict__ f1,
    const float* __restrict__ cf,
    const float* __restrict__ W1, const float* __restrict__ b1,
    const float* __restrict__ W2, const float* __restrict__ b2,
    float* __restrict__ out) {
  __shared__ unsigned int  s_cnt[64];
  __shared__ float         s_d2[64 * CAP_];
  __shared__ unsigned int  s_idx[64 * CAP_];
  __shared__ unsigned char s_flag[64 * CAP_];

  const int tid  = threadIdx.x;
  const int wav  = tid >> 5;        // wave id 0..3 (wave32)
  const int lane = tid & 31;
  const int li   = lane & 15;       // column / M-index within tile
  const int hh   = lane >> 4;       // lane half selects K pair / M half

  const int bq = blockIdx.x * 64;   // 64 queries per block
  const int b  = bq / N0_;
  const int q0 = bq % N0_;
  const int qt = q0 + wav * 16;     // this wave's query tile base

  const float* tg  = targ4 + (size_t)b * N0_ * 4;
  const float* pp  = p14   + (size_t)b * N1_ * 4;
  const float* qnb = qn    + (size_t)b * N0_;

  if (lane < 16) s_cnt[wav * 16 + lane] = 0;   // own-wave region, DS in-order

  // ---- per-lane resident weights: lane owns channels h = 4*lane..4*lane+3
  float w1r[6][4], b1r[4], w2r[4][3], w2t[3][3], b2r[3];
#pragma unroll
  for (int i = 0; i < 6; ++i)
#pragma unroll
    for (int t = 0; t < 4; ++t) w1r[i][t] = W1[i * H_ + 4 * lane + t];
#pragma unroll
  for (int t = 0; t < 4; ++t) b1r[t] = b1[4 * lane + t];
#pragma unroll
  for (int t = 0; t < 4; ++t)
#pragma unroll
    for (int k = 0; k < 3; ++k) w2r[t][k] = W2[(4 * lane + t) * 3 + k];
#pragma unroll
  for (int i = 0; i < 3; ++i)
#pragma unroll
    for (int k = 0; k < 3; ++k) w2t[i][k] = W2[(H_ + i) * 3 + k];
#pragma unroll
  for (int k = 0; k < 3; ++k) b2r[k] = b2[k];

  // ---- A fragment (constant over the candidate scan)
  // A 16x4 layout: lane L: M = L&15, VGPR r, K = r + 2*(L>>4)
  const int q_lane = qt + li;
  v4f tq4 = *(const v4f*)(tg + (size_t)q_lane * 4);
  v2f afrag;
  afrag.x = hh ? tq4.z : tq4.x;
  afrag.y = hh ? tq4.w : tq4.y;     // tq4.w == 0 -> K=3 slot exact zero

  float qnr[8];
#pragma unroll
  for (int r = 0; r < 8; ++r) qnr[r] = qnb[qt + r + 8 * hh];

  // ================= Phase 1: pipelined WMMA distance scan =================
  v4f pjc = *(const v4f*)(pp + (size_t)li * 4);       // tile 0 pre-issued
  for (int jb = 0; jb < N1_; jb += 16) {
    // issue next tile's load + a deep prefetch before consuming the current
    const int jn = (jb + 16) & (N1_ - 1);
    v4f pjn = *(const v4f*)(pp + (size_t)(jn + li) * 4);
    __builtin_prefetch(pp + (size_t)(((jb + 256) & (N1_ - 1)) + li) * 4, 0, 3);

    v2f bfrag;                      // B 4x16: lane L: N=L&15, K = r + 2*(L>>4)
    bfrag.x = hh ? pjc.z : pjc.x;
    bfrag.y = hh ? pjc.w : pjc.y;   // pjc.w = |p|^2, multiplied by A's 0
    const float cn = pjc.w;

    v8f cz = {};
    v8f d = __builtin_amdgcn_wmma_f32_16x16x4_f32(
        false, afrag, false, bfrag, (short)0, cz, false, false);

    float d2v[8];
#pragma unroll
    for (int r = 0; r < 8; ++r) d2v[r] = qnr[r] + cn - 2.0f * d[r];

    // cheap "any hit" test: min-tree + single compare (common path = skip)
    float m0 = fminf(fminf(d2v[0], d2v[1]), fminf(d2v[2], d2v[3]));
    float m1 = fminf(fminf(d2v[4], d2v[5]), fminf(d2v[6], d2v[7]));
    if (fminf(m0, m1) < R2_) {      // rare path: sparse neighborhoods
      const int j = jb + li;        // lane's candidate column
#pragma unroll
      for (int r = 0; r < 8; ++r) {
        if (d2v[r] < R2_) {
          const int m = r + 8 * hh; // query row within this wave's tile
          unsigned pos = atomicAdd(&s_cnt[wav * 16 + m], 1u);
          if (pos < CAP_) {
            s_d2[(wav * 16 + m) * CAP_ + pos]  = d2v[r];
            s_idx[(wav * 16 + m) * CAP_ + pos] = (unsigned)j;
          }
        }
      }
    }
    pjc = pjn;
  }

  // ================= Phase 2: select + MLP + project =================
  const size_t ob = (size_t)b * N0_;
  for (int m = 0; m < 16; ++m) {
    const int q = qt + m;
    unsigned c = s_cnt[wav * 16 + m];
    if (c > CAP_) c = CAP_;
    if (c == 0) {                   // no neighbor: residual == 0
      if (lane < 3) out[(ob + q) * 3 + lane] = cf[(ob + q) * 3 + lane];
      continue;
    }
    const int base   = (wav * 16 + m) * CAP_;
    const bool useFl = (c > K_);    // statistically never, exact when it is
    if (useFl && lane == 0) {
      for (unsigned e = 0; e < c; ++e) {
        float de = s_d2[base + e];
        int rank = 0;
        for (unsigned jj = 0; jj < c; ++jj) {
          float dj = s_d2[base + jj];
          rank += (dj < de) || (dj == de && jj < e);
        }
        s_flag[base + e] = (rank < K_) ? 1 : 0;
      }
    }

    v4f tq = *(const v4f*)(tg + (size_t)q * 4);
    float pooled[4] = {-3.402823466e38f, -3.402823466e38f,
                       -3.402823466e38f, -3.402823466e38f};
    for (unsigned e = 0; e < c; ++e) {
      if (useFl && !s_flag[base + e]) continue;
      const unsigned j = s_idx[base + e];
      v4f pj = *(const v4f*)(pp + (size_t)j * 4);
      const float* f1p = f1 + ((size_t)b * N1_ + j) * 3;
      const float x0 = pj.x - tq.x, x1 = pj.y - tq.y, x2 = pj.z - tq.z;
      const float x3 = f1p[0], x4 = f1p[1], x5 = f1p[2];
#pragma unroll
      for (int t = 0; t < 4; ++t) {
        float a = b1r[t];
        a = fmaf(x0, w1r[0][t], a);
        a = fmaf(x1, w1r[1][t], a);
        a = fmaf(x2, w1r[2][t], a);
        a = fmaf(x3, w1r[3][t], a);
        a = fmaf(x4, w1r[4][t], a);
        a = fmaf(x5, w1r[5][t], a);
        a = fmaxf(a, 0.0f);
        pooled[t] = fmaxf(pooled[t], a);
      }
    }

    float part[3];
#pragma unroll
    for (int k = 0; k < 3; ++k)
      part[k] = pooled[0] * w2r[0][k] + pooled[1] * w2r[1][k] +
                pooled[2] * w2r[2][k] + pooled[3] * w2r[3][k];
    if (lane == 0) {
      const float* f0p = f0 + (ob + q) * 3;
#pragma unroll
      for (int k = 0; k < 3; ++k)
        part[k] += f0p[0] * w2t[0][k] + f0p[1] * w2t[1][k] +
                   f0p[2] * w2t[2][k] + b2r[k];
    }
#pragma unroll
    for (int off = 16; off > 0; off >>= 1) {
#pragma unroll
      for (int k = 0; k < 3; ++k) part[k] += __shfl_xor(part[k], off, 32);
    }
    if (lane == 0) {
#pragma unroll
      for (int k = 0; k < 3; ++k)
        out[(ob + q) * 3 + k] = cf[(ob + q) * 3 + k] + part[k];
    }
  }
}

// ---------------------------------------------------------------------------
extern "C" void kernel_launch(void* const* d_in, const int* in_sizes, int n_in,
                              void* d_out, int out_size, void* d_ws,
                              size_t ws_size, hipStream_t stream) {
  (void)in_sizes; (void)n_in; (void)out_size; (void)ws_size;
  const float* p0 = (const float*)d_in[0];
  const float* p1 = (const float*)d_in[1];
  const float* f0 = (const float*)d_in[2];
  const float* f1 = (const float*)d_in[3];
  const float* cf = (const float*)d_in[4];
  const float* W1 = (const float*)d_in[5];
  const float* b1 = (const float*)d_in[6];
  const float* W2 = (const float*)d_in[7];
  const float* b2 = (const float*)d_in[8];
  float* out = (float*)d_out;

  float* targ4 = (float*)d_ws;                    // B*N0*4 floats
  float* p14   = targ4 + (size_t)B_ * N0_ * 4;    // B*N1*4 floats
  float* qnv   = p14 + (size_t)B_ * N1_ * 4;      // B*N0 floats

  const int npre = B_ * N0_;
  ff3d_pre<<<(npre + 255) / 256, 256, 0, stream>>>(p0, p1, cf, targ4, p14, qnv);

  const int nblk = (B_ * N0_) / 64;               // 256 blocks x 128 threads
  ff3d_main<<<nblk, 128, 0, stream>>>(targ4, p14, qnv, f0, f1, cf,
                                      W1, b1, W2, b2, out);
}